// EISepLSTM_20607253086666
// MI455X (gfx1250) — compile-verified
//
#include <hip/hip_runtime.h>
#include <hip/hip_bf16.h>
#include <math.h>
#include <stdint.h>

// ---------------------------------------------------------------------------
// EISepLSTM on gfx1250 (MI455X): f32 WMMA (V_WMMA_F32_16X16X4_F32) everywhere,
// with GLOBAL_LOAD_ASYNC_TO_LDS_B128 double-buffered staging in the GEMMs.
// Dims: T=512, B=128, I=H=O=512.
// ---------------------------------------------------------------------------

typedef float v2f __attribute__((ext_vector_type(2)));
typedef float v8f __attribute__((ext_vector_type(8)));

#define TT 512
#define BB 128
#define HH 512

__device__ __forceinline__ void async_load_b128(uint32_t lds_off, const void* gaddr)
{
    // GLOBAL_LOAD_ASYNC_TO_LDS_B128: VDST = LDS byte address, VADDR = 64-bit
    // global address, saddr = off. Tracked by ASYNCcnt.
    asm volatile("global_load_async_to_lds_b128 %0, %1, off"
                 :: "v"(lds_off), "v"(gaddr) : "memory");
}

__device__ __forceinline__ void wait_asynccnt0()
{
    asm volatile("s_wait_asynccnt 0" ::: "memory");
}

__device__ __forceinline__ uint32_t lds_addr_of(const void* shared_ptr)
{
    // Generic pointers into the LDS aperture carry the LDS byte offset in
    // their low 32 bits.
    return (uint32_t)(uintptr_t)shared_ptr;
}

// ---------------------------------------------------------------------------
// Transpose (+optional ReLU) a 512x512 weight: out[k*512+n] = f(in[n*512+k])
// ---------------------------------------------------------------------------
__global__ __launch_bounds__(256) void transpose512_kernel(
    const float* __restrict__ in, float* __restrict__ out, int do_relu)
{
    int idx = blockIdx.x * 256 + threadIdx.x;   // 0 .. 512*512-1
    int k = idx >> 9;
    int n = idx & 511;
    float v = in[n * 512 + k];
    if (do_relu) v = fmaxf(v, 0.0f);
    out[idx] = v;
}

// ---------------------------------------------------------------------------
// GEMM: C[m, n] = sum_k A[m,k] * BT[k,n] + bias[n]
//   A:  65536 x 512 row-major, BT: 512 x 512 (k-major).
// Block: 256 threads (8 waves). Block tile 128M x 64N.
// Both A slab (128x32, LDS stride 36 -> conflict-free 16x4 fragments) and
// B slab (32x64) are staged via async global->LDS B128 copies, double-buffered
// so the 8 WMMA k-steps of slab s overlap the fetch of slab s+1.
// All 8 waves share the B slab (same 64 N-columns) -> 8x less global traffic.
// ---------------------------------------------------------------------------
__global__ __launch_bounds__(256) void gemm_f32_wmma_kernel(
    const float* __restrict__ A,
    const float* __restrict__ BT,
    const float* __restrict__ bias,
    float* __restrict__ C)
{
    __shared__ float a_lds[2][128 * 36];
    __shared__ float b_lds[2][32 * 64];

    const int tid   = threadIdx.x;
    const int wave  = tid >> 5;
    const int lane  = tid & 31;
    const int row16 = lane & 15;
    const int kb    = (lane >> 4) << 1;   // K sub-offset per ISA A/B layout
    const int mBase = (lane >> 4) * 8;    // C/D layout: lanes>=16 hold M=8..15

    const size_t m0 = (size_t)blockIdx.y * 128;
    const int    n0 = blockIdx.x * 64;

    v8f acc[4];
#pragma unroll
    for (int j = 0; j < 4; ++j)
#pragma unroll
        for (int e = 0; e < 8; ++e) acc[j][e] = 0.0f;

    // ---- async staging of slab s into buffer buf ----
    auto issue_slab = [&](int s, int buf) {
        const int k0 = s * 32;
        const uint32_t abase = lds_addr_of(&a_lds[buf][0]);
        const uint32_t bbase = lds_addr_of(&b_lds[buf][0]);
        // A slab: 128 rows x 32 floats = 1024 x 16B chunks, 4 per thread
#pragma unroll
        for (int i = 0; i < 4; ++i) {
            int c  = i * 256 + tid;
            int r  = c >> 3;
            int cc = (c & 7) << 2;
            async_load_b128(abase + (uint32_t)(r * 36 + cc) * 4u,
                            A + (m0 + r) * 512 + k0 + cc);
        }
        // B slab: 32 rows x 64 floats = 512 x 16B chunks, 2 per thread
#pragma unroll
        for (int i = 0; i < 2; ++i) {
            int c  = i * 256 + tid;
            int r  = c >> 4;
            int cc = (c & 15) << 2;
            async_load_b128(bbase + (uint32_t)(r * 64 + cc) * 4u,
                            BT + (size_t)(k0 + r) * 512 + n0 + cc);
        }
    };

    issue_slab(0, 0);

    const int arow = wave * 16 + row16;
    for (int s = 0; s < 16; ++s) {
        const int cur = s & 1;
        wait_asynccnt0();        // this wave's slab-s LDS writes have landed
        __syncthreads();         // ...and everyone else's too; prior reads done
        if (s + 1 < 16) issue_slab(s + 1, cur ^ 1);

#pragma unroll
        for (int ks = 0; ks < 8; ++ks) {
            const int kk = ks * 4 + kb;
            v2f a;
            a.x = a_lds[cur][arow * 36 + kk];
            a.y = a_lds[cur][arow * 36 + kk + 1];
#pragma unroll
            for (int j = 0; j < 4; ++j) {
                const int nl = j * 16 + row16;
                v2f b;
                b.x = b_lds[cur][kk * 64 + nl];
                b.y = b_lds[cur][(kk + 1) * 64 + nl];
                acc[j] = __builtin_amdgcn_wmma_f32_16x16x4_f32(
                    false, a, false, b, (short)0, acc[j], false, false);
            }
        }
    }

    // epilogue: + bias, store per C/D VGPR layout
#pragma unroll
    for (int j = 0; j < 4; ++j) {
        const int   n  = n0 + j * 16 + row16;
        const float bn = bias[n];
#pragma unroll
        for (int r = 0; r < 8; ++r) {
            size_t m = m0 + wave * 16 + mBase + r;
            C[m * 512 + n] = acc[j][r] + bn;
        }
    }
}

// ---------------------------------------------------------------------------
// Recurrent scan. Batch rows are independent across the recurrence, so each
// of 8 blocks owns 16 batch rows for all 512 steps; its h tile (16x512) lives
// in LDS (stride 516 -> conflict-free 16x4 A fragments). ei_t computed inline:
// softmax2 difference == tanh(((l0+g0)-(l1+g1))/(2*tau)).
// Per step: (h*ei) @ WrecT via f32 WMMA, then softplus epilogue; h_t overwrites
// ig_t in-place in the global ig/hs buffer.
// ---------------------------------------------------------------------------
__global__ __launch_bounds__(256) void eiseplstm_scan_kernel(
    const float* __restrict__ gumbel_u,   // (T, H, 2)
    const float* __restrict__ logit,      // (H, 2)
    const float* __restrict__ WrecT,      // (512, 512) relu'd, k-major
    const float* __restrict__ rec_bias,   // (H)
    float* __restrict__ ig_hs,            // (T, B, H): in = ig, out = hs
    float* __restrict__ h_last,           // (B, H)
    float* __restrict__ c_last)           // (B, H) -> zeros
{
    __shared__ float h_lds[16 * 516];
    __shared__ float ei_lds[512];

    const int tid   = threadIdx.x;
    const int wave  = tid >> 5;
    const int lane  = tid & 31;
    const int row16 = lane & 15;
    const int kb    = (lane >> 4) << 1;
    const int mBase = (lane >> 4) * 8;
    const int bblk  = blockIdx.x;          // batch rows [bblk*16, bblk*16+16)

    for (int i = tid; i < 16 * 516; i += 256) h_lds[i] = 0.0f;
    __syncthreads();

    for (int t = 0; t < TT; ++t) {
        // ---- ei_t (gumbel-softmax difference, exact tanh form) ----
        for (int i = tid; i < HH; i += 256) {
            float u0 = gumbel_u[((size_t)t * HH + i) * 2 + 0];
            float u1 = gumbel_u[((size_t)t * HH + i) * 2 + 1];
            float g0 = -logf(-logf(u0 + 1e-10f) + 1e-10f);
            float g1 = -logf(-logf(u1 + 1e-10f) + 1e-10f);
            float d  = ((logit[i * 2] + g0) - (logit[i * 2 + 1] + g1)) * 5.0f; // 1/(2*0.1)
            ei_lds[i] = tanhf(d);
        }
        __syncthreads();

        // ---- rec = (h * ei_t) @ WrecT ----
        v8f acc[4];
#pragma unroll
        for (int j = 0; j < 4; ++j)
#pragma unroll
            for (int e = 0; e < 8; ++e) acc[j][e] = 0.0f;

        for (int k = 0; k < HH; k += 4) {
            const int kk = k + kb;
            v2f a;
            a.x = h_lds[row16 * 516 + kk]     * ei_lds[kk];
            a.y = h_lds[row16 * 516 + kk + 1] * ei_lds[kk + 1];
#pragma unroll
            for (int j = 0; j < 4; ++j) {
                const int n = wave * 64 + j * 16 + row16;
                v2f b;
                b.x = WrecT[(size_t)kk * 512 + n];
                b.y = WrecT[(size_t)(kk + 1) * 512 + n];
                acc[j] = __builtin_amdgcn_wmma_f32_16x16x4_f32(
                    false, a, false, b, (short)0, acc[j], false, false);
            }
        }
        __syncthreads();   // all h_lds / ei_lds reads done

        // ---- h_new = softplus(ig + rec + rec_bias); store hs and LDS h ----
#pragma unroll
        for (int j = 0; j < 4; ++j) {
            const int   n  = wave * 64 + j * 16 + row16;
            const float rb = rec_bias[n];
#pragma unroll
            for (int r = 0; r < 8; ++r) {
                const int mloc = mBase + r;
                size_t gidx = (((size_t)t * BB) + bblk * 16 + mloc) * (size_t)HH + n;
                float xv = acc[j][r] + ig_hs[gidx] + rb;
                // log1p(exp(x)) stable form (matches jax.nn.softplus)
                float hv = fmaxf(xv, 0.0f) + log1pf(expf(-fabsf(xv)));
                ig_hs[gidx] = hv;
                h_lds[mloc * 516 + n] = hv;
            }
        }
        __syncthreads();
    }

    // ---- h_last / c_last ----
    for (int i = tid; i < 16 * HH; i += 256) {
        int r = i >> 9, n = i & 511;
        size_t o = ((size_t)bblk * 16 + r) * (size_t)HH + n;
        h_last[o] = h_lds[r * 516 + n];
        c_last[o] = 0.0f;
    }
}

// ---------------------------------------------------------------------------
// Host-side launch
// ---------------------------------------------------------------------------
extern "C" void kernel_launch(void* const* d_in, const int* in_sizes, int n_in,
                              void* d_out, int out_size, void* d_ws, size_t ws_size,
                              hipStream_t stream)
{
    const float* x      = (const float*)d_in[0];   // (T,B,I)
    const float* gu     = (const float*)d_in[1];   // (T,H,2)
    const float* W_in   = (const float*)d_in[2];   // (H,I)
    const float* b_in   = (const float*)d_in[3];   // (H)
    const float* Wrec   = (const float*)d_in[4];   // (H,H)
    const float* rbias  = (const float*)d_in[5];   // (H)
    const float* logit  = (const float*)d_in[6];   // (H,2)
    const float* W_out  = (const float*)d_in[7];   // (O,H)
    const float* b_out  = (const float*)d_in[8];   // (O)

    float* ws    = (float*)d_ws;
    float* WinT  = ws;                 // 512*512
    float* WrecT = ws + 262144;        // 512*512
    float* WoutT = ws + 524288;        // 512*512
    float* ig_hs = ws + 786432;        // T*B*H = 33554432 floats

    float* out    = (float*)d_out;                       // (T,B,O)
    float* h_last = out + (size_t)TT * BB * HH;          // (B,H)
    float* c_last = h_last + (size_t)BB * HH;            // (B,H)

    // 1) transpose weights (ReLU applied to Wrec)
    transpose512_kernel<<<1024, 256, 0, stream>>>(W_in,  WinT,  0);
    transpose512_kernel<<<1024, 256, 0, stream>>>(Wrec,  WrecT, 1);
    transpose512_kernel<<<1024, 256, 0, stream>>>(W_out, WoutT, 0);

    // 2) ig = x @ W_in^T + b_in   (M = T*B = 65536)
    gemm_f32_wmma_kernel<<<dim3(8, 512), 256, 0, stream>>>(x, WinT, b_in, ig_hs);

    // 3) recurrent scan (hs overwrites ig in place; also h_last, c_last)
    eiseplstm_scan_kernel<<<8, 256, 0, stream>>>(gu, logit, WrecT, rbias,
                                                 ig_hs, h_last, c_last);

    // 4) outputs = hs @ W_out^T + b_out
    gemm_f32_wmma_kernel<<<dim3(8, 512), 256, 0, stream>>>(ig_hs, WoutT, b_out, out);
}